// Aggregator_19670950216024
// MI455X (gfx1250) — compile-verified
//
#include <hip/hip_runtime.h>
#include <hip/hip_bf16.h>

typedef float v2f __attribute__((ext_vector_type(2)));
typedef float v8f __attribute__((ext_vector_type(8)));

#define DIM      128
#define TILE_M   16
#define XPAD     132          // padded LDS row stride (floats): 132 % 64 = 4 -> conflict-free
#define WAVES_PER_BLOCK 4

// ---------------------------------------------------------------------------
// Kernel 1: x = ego  (re-initialize scratch accumulator every launch)
// ---------------------------------------------------------------------------
__global__ __launch_bounds__(256) void gcn_init_kernel(const float4* __restrict__ ego,
                                                       float4* __restrict__ x,
                                                       int n4) {
    int i = blockIdx.x * 256 + threadIdx.x;
    if (i < n4) x[i] = ego[i];
}

// ---------------------------------------------------------------------------
// Kernel 2: x[dst, :] += w * ego[src, :]   (COO scatter, 128 threads/edge)
// ---------------------------------------------------------------------------
__global__ __launch_bounds__(256) void gcn_scatter_kernel(const float* __restrict__ ego,
                                                          const long long* __restrict__ edge_index,
                                                          const float* __restrict__ edge_weight,
                                                          float* __restrict__ x,
                                                          int num_edges) {
    long long idx = (long long)blockIdx.x * 256 + threadIdx.x;
    int e = (int)(idx >> 7);          // 128 threads per edge
    int d = (int)(idx & 127);
    if (e >= num_edges) return;
    long long dst = edge_index[e];                 // row 0 = dst
    long long src = edge_index[(long long)num_edges + e]; // row 1 = src
    float w = edge_weight[e];
    float v = ego[src * DIM + d] * w;
    atomicAdd(x + dst * DIM + d, v);               // -> global_atomic_add_f32 (L2-resident)
}

// ---------------------------------------------------------------------------
// Kernel 3: out = LeakyReLU(x @ W^T + b), fp32 WMMA 16x16x4, one wave / 16 rows
// ---------------------------------------------------------------------------
__global__ __launch_bounds__(WAVES_PER_BLOCK * 32) void gcn_gemm_kernel(
        const float* __restrict__ X,     // [N, 128] = ego + side
        const float* __restrict__ W,     // [128, 128] row-major; B[k][n] = W[n][k]
        const float* __restrict__ bias,  // [128]
        float* __restrict__ Y,           // [N, 128]
        int n_tiles) {
    __shared__ float xs[WAVES_PER_BLOCK][TILE_M * XPAD];

    const int wave = threadIdx.x >> 5;
    const int lane = threadIdx.x & 31;
    const int tile = blockIdx.x * WAVES_PER_BLOCK + wave;
    if (tile >= n_tiles) return;      // wave-uniform; no cross-wave LDS sharing
    const int row0 = tile * TILE_M;

    float* xw = xs[wave];

    // ---- stage X tile (16 x 128 f32) into LDS, coalesced float4 per row ----
    {
        const float4* Xg = reinterpret_cast<const float4*>(X + (long long)row0 * DIM);
#pragma unroll
        for (int r = 0; r < TILE_M; ++r) {
            float4 v = Xg[r * (DIM / 4) + lane];
            *reinterpret_cast<float4*>(&xw[r * XPAD + lane * 4]) = v;
        }
    }

    // fragment coordinates (ISA 7.12.2, 32-bit A 16x4 / C-D 16x16 layouts)
    const int m  = lane & 15;   // A: row; B: column n (within col-tile)
    const int kh = lane >> 4;   // 0 -> K{0,1}, 1 -> K{2,3}

    v8f acc[8];
#pragma unroll
    for (int ct = 0; ct < 8; ++ct)
#pragma unroll
        for (int r = 0; r < 8; ++r) acc[ct][r] = 0.0f;

    // ---- K loop: 32 steps of K=4, 8 column tiles per step ----
    for (int kk = 0; kk < DIM; kk += 4) {
        const int ko = kk + 2 * kh;                       // even -> 8B aligned
        v2f a = *reinterpret_cast<const v2f*>(&xw[m * XPAD + ko]);  // ds_load_b64
#pragma unroll
        for (int ct = 0; ct < 8; ++ct) {
            // B fragment: lane holds B[ko..ko+1][n] = W[n][ko..ko+1], n = ct*16 + m
            v2f bf = *reinterpret_cast<const v2f*>(W + (ct * 16 + m) * DIM + ko);
            acc[ct] = __builtin_amdgcn_wmma_f32_16x16x4_f32(
                /*neg_a=*/false, a, /*neg_b=*/false, bf,
                /*c_mod=*/(short)0, acc[ct],
                /*reuse_a=*/false, /*reuse_b=*/false);
        }
    }

    // ---- epilogue: bias + LeakyReLU(0.01), D layout: M = r + 8*kh, N = ct*16 + m
#pragma unroll
    for (int ct = 0; ct < 8; ++ct) {
        const int n = ct * 16 + m;
        const float bn = bias[n];
#pragma unroll
        for (int r = 0; r < 8; ++r) {
            const int mm = r + 8 * kh;
            float h = acc[ct][r] + bn;
            Y[(long long)(row0 + mm) * DIM + n] = (h > 0.0f) ? h : 0.01f * h;
        }
    }
}

// ---------------------------------------------------------------------------
extern "C" void kernel_launch(void* const* d_in, const int* in_sizes, int n_in,
                              void* d_out, int out_size, void* d_ws, size_t ws_size,
                              hipStream_t stream) {
    const float*     ego = (const float*)d_in[0];      // [N,128] f32
    const long long* ei  = (const long long*)d_in[1];  // [2,E]   i64
    const float*     ew  = (const float*)d_in[2];      // [E]     f32
    const float*     W   = (const float*)d_in[3];      // [128,128]
    const float*     b   = (const float*)d_in[4];      // [128]
    float*           out = (float*)d_out;              // [N,128] f32
    float*           x   = (float*)d_ws;               // [N,128] scratch accumulator

    const int N = in_sizes[0] / DIM;       // 100000
    const int E = in_sizes[1] / 2;         // 1600000

    // 1) x = ego
    {
        int n4 = (N * DIM) / 4;
        int grid = (n4 + 255) / 256;
        gcn_init_kernel<<<grid, 256, 0, stream>>>(
            (const float4*)ego, (float4*)x, n4);
    }
    // 2) x[dst] += w * ego[src]
    {
        long long total = (long long)E * DIM;
        int grid = (int)((total + 255) / 256);
        gcn_scatter_kernel<<<grid, 256, 0, stream>>>(ego, ei, ew, x, E);
    }
    // 3) out = LeakyReLU(x @ W^T + b)
    {
        int n_tiles = N / TILE_M;          // 6250 (N divisible by 16)
        int grid = (n_tiles + WAVES_PER_BLOCK - 1) / WAVES_PER_BLOCK;
        gcn_gemm_kernel<<<grid, WAVES_PER_BLOCK * 32, 0, stream>>>(x, W, b, out, n_tiles);
    }
}